// MuyGPLayer_85272280695391
// MI455X (gfx1250) — compile-verified
//
#include <hip/hip_runtime.h>

typedef __attribute__((ext_vector_type(2))) float v2f;
typedef __attribute__((ext_vector_type(8))) float v8f;

#ifndef __has_builtin
#define __has_builtin(x) 0
#endif
#if __has_builtin(__builtin_amdgcn_wmma_f32_16x16x4_f32)
#define HAVE_WMMA_F32 1
#else
#define HAVE_WMMA_F32 0
#endif

#if defined(__gfx1250__)
#define HAVE_ASYNC_LDS 1
#else
#define HAVE_ASYNC_LDS 0
#endif

static constexpr int Bq    = 1024;
static constexpr int N     = 100000;
static constexpr int D     = 128;
static constexpr int O     = 29;
static constexpr int K     = 128;
static constexpr int NPAD  = 100096;           // 782*128 == 256*391
static constexpr int SLAB  = 128;              // query rows per distance slab
static constexpr int NSLAB = Bq / SLAB;        // 8
static constexpr int CHUNK = NPAD / 256;       // 391, per-thread range in select
static constexpr float FMAXV = 3.402823466e+38f;

// ---- async global->LDS (CDNA5, ASYNCcnt-tracked, bypasses VGPRs) ----------
__device__ __forceinline__ void async_load_b128(const void* gptr, void* lptr) {
#if HAVE_ASYNC_LDS
    // generic shared pointer: low 32 bits are the LDS byte offset (aperture rule)
    unsigned lds = (unsigned)(unsigned long long)lptr;
    unsigned long long ga = (unsigned long long)gptr;
    asm volatile("global_load_async_to_lds_b128 %0, %1, off"
                 :: "v"(lds), "v"(ga) : "memory");
#else
    (void)gptr; (void)lptr;
#endif
}
__device__ __forceinline__ void wait_async0() {
#if HAVE_ASYNC_LDS
    asm volatile("s_wait_asynccnt 0" ::: "memory");
#endif
}

// ---------------------------------------------------------------- norms ----
__global__ __launch_bounds__(256) void norms_kernel(const float* __restrict__ src,
                                                    float* __restrict__ out, int rows) {
    const int lane = threadIdx.x & 31, wv = threadIdx.x >> 5;
    const int row = blockIdx.x * 8 + wv;
    if (row >= rows) return;
    const float* p = src + (size_t)row * D;
    float s = 0.f;
    for (int d = lane; d < D; d += 32) { float v = p[d]; s += v * v; }
    for (int off = 16; off > 0; off >>= 1) s += __shfl_xor(s, off, 32);
    if (lane == 0) out[row] = s;
}

// ---------------------------------------------------------------- ymean ----
__global__ __launch_bounds__(32) void ymean_kernel(const float* __restrict__ x,
                                                   const float* __restrict__ W,
                                                   const float* __restrict__ bias,
                                                   float* __restrict__ ymean) {
    const int b = blockIdx.x, o = threadIdx.x;
    if (o >= O) return;
    const float* xr = x + (size_t)b * D;
    const float* wr = W + (size_t)o * D;
    float s = bias[o];
    for (int d = 0; d < D; ++d) s += xr[d] * wr[d];
    ymean[b * O + o] = s;
}

// -------------------------------------------- distance tiles (WMMA f32) ----
// grid (NPAD/64, SLAB/16), block 128 (4 waves). Wave w computes a 16x16 tile
// of dot(x_row, trainX_col) via V_WMMA_F32_16X16X4_F32, 32 k-steps.
__global__ __launch_bounds__(128) void dist_kernel(const float* __restrict__ x,
                                                   const float* __restrict__ trainX,
                                                   const float* __restrict__ xn,
                                                   const float* __restrict__ tn,
                                                   float* __restrict__ distSlab,
                                                   int rowBase) {
    constexpr int S = D + 4;                       // LDS row stride (bank-conflict pad)
    __shared__ __align__(16) float xs[16 * S];
    __shared__ __align__(16) float ts[64 * S];
    __shared__ float xnl[16], tnl[64];
    const int tid = threadIdx.x;
    const int lane = tid & 31, wv = tid >> 5;
    const int r0 = rowBase + blockIdx.y * 16;
    const int j0 = blockIdx.x * 64;

#if HAVE_ASYNC_LDS
    for (int q = tid; q < 16 * (D / 4); q += 128) {          // 16 query rows
        const int r = q >> 5, c4 = q & 31;
        async_load_b128(x + (size_t)(r0 + r) * D + c4 * 4, &xs[r * S + c4 * 4]);
    }
    for (int q = tid; q < 64 * (D / 4); q += 128) {          // 64 candidate rows
        const int r = q >> 5, c4 = q & 31;
        const int j = min(j0 + r, N - 1);                    // clamp; bad cols masked later
        async_load_b128(trainX + (size_t)j * D + c4 * 4, &ts[r * S + c4 * 4]);
    }
#else
    for (int q = tid; q < 16 * (D / 4); q += 128) {
        int r = q >> 5, c4 = q & 31;
        float4 v = ((const float4*)(x + (size_t)(r0 + r) * D))[c4];
        *((float4*)&xs[r * S + c4 * 4]) = v;
    }
    for (int q = tid; q < 64 * (D / 4); q += 128) {
        int r = q >> 5, c4 = q & 31;
        int j = min(j0 + r, N - 1);
        float4 v = ((const float4*)(trainX + (size_t)j * D))[c4];
        *((float4*)&ts[r * S + c4 * 4]) = v;
    }
#endif
    if (tid < 16) xnl[tid] = xn[r0 + tid];
    if (tid < 64) tnl[tid] = (j0 + tid < N) ? tn[j0 + tid] : 0.f;
    wait_async0();
    __syncthreads();

    const int n0 = wv * 16;
    const int m  = lane & 15;
    const int k0 = (lane >> 4) * 2;
    const int hi = lane >> 4, nn = lane & 15;
    (void)m; (void)k0; (void)hi; (void)nn;
#if HAVE_WMMA_F32
    v8f acc = {0.f, 0.f, 0.f, 0.f, 0.f, 0.f, 0.f, 0.f};
    for (int s = 0; s < 32; ++s) {
        const int kk = 4 * s + k0;
        v2f a2 = *(const v2f*)&xs[m * S + kk];                // A: M=m, K=kk,kk+1
        v2f b2 = *(const v2f*)&ts[(n0 + m) * S + kk];         // B: N=m, K=kk,kk+1
        acc = __builtin_amdgcn_wmma_f32_16x16x4_f32(false, a2, false, b2,
                                                    (short)0, acc, false, false);
    }
#else
    float acc[8] = {0.f, 0.f, 0.f, 0.f, 0.f, 0.f, 0.f, 0.f};
    for (int k = 0; k < D; ++k) {
        float tv = ts[(n0 + nn) * S + k];
        #pragma unroll
        for (int vv = 0; vv < 8; ++vv) acc[vv] += xs[(vv + 8 * hi) * S + k] * tv;
    }
#endif
    #pragma unroll
    for (int vv = 0; vv < 8; ++vv) {                          // C layout: M=vv+8*hi, N=nn
        const int M = vv + 8 * hi;
        const int j = j0 + n0 + nn;
        float d2 = (j < N) ? fmaxf(xnl[M] + tnl[n0 + nn] - 2.f * acc[vv], 0.f) : FMAXV;
        distSlab[(size_t)(blockIdx.y * 16 + M) * NPAD + j] = d2;
    }
}

// --------------------------------- deterministic radix-select of K=128 -----
__global__ __launch_bounds__(256) void select_kernel(const float* __restrict__ distSlab,
                                                     int rowBase, int* __restrict__ idxOut) {
    __shared__ int hist[256];
    __shared__ unsigned sPrefix;
    __shared__ int sWant, sCLt;
    __shared__ int ltc[256], eqc[256], ltoff[256], eqoff[256];
    const int tid = threadIdx.x;
    const float* row = distSlab + (size_t)blockIdx.x * NPAD;

    unsigned prefix = 0; int want = K;                         // rank (1-indexed) of kth smallest
    for (int shift = 24; shift >= 0; shift -= 8) {
        hist[tid] = 0;
        __syncthreads();
        for (int i = tid; i < NPAD; i += 256) {
            unsigned u = __float_as_uint(row[i]);
            bool ok = (shift == 24) || ((u >> (shift + 8)) == prefix);
            if (ok) atomicAdd(&hist[(u >> shift) & 255], 1);
        }
        __syncthreads();
        if (tid == 0) {
            int run = 0, bsel = 255, wn = want;
            for (int bb = 0; bb < 256; ++bb) {
                int c = hist[bb];
                if (run + c >= want) { bsel = bb; wn = want - run; break; }
                run += c;
            }
            sPrefix = (prefix << 8) | (unsigned)bsel;
            sWant = wn;
        }
        __syncthreads();
        prefix = sPrefix; want = sWant;
        __syncthreads();
    }
    const unsigned kbits = prefix;                             // bits of kth-smallest value

    // stable compaction: thread t owns contiguous range [t*CHUNK, (t+1)*CHUNK)
    const int beg = tid * CHUNK, end = beg + CHUNK;
    int myLt = 0, myEq = 0;
    for (int i = beg; i < end; ++i) {
        unsigned u = __float_as_uint(row[i]);
        myLt += (u < kbits); myEq += (u == kbits);
    }
    ltc[tid] = myLt; eqc[tid] = myEq;
    __syncthreads();
    if (tid == 0) {
        int rl = 0, re = 0;
        for (int t = 0; t < 256; ++t) { ltoff[t] = rl; rl += ltc[t]; eqoff[t] = re; re += eqc[t]; }
        sCLt = rl;
    }
    __syncthreads();
    int lpos = ltoff[tid], epos = sCLt + eqoff[tid];
    int* out = idxOut + (size_t)(rowBase + blockIdx.x) * K;
    for (int i = beg; i < end; ++i) {
        unsigned u = __float_as_uint(row[i]);
        if (u < kbits) out[lpos++] = i;
        else if (u == kbits) { if (epos < K) out[epos] = i; epos++; }
    }
}

// ---------------- per-batch: gather nX, auto = kernel(nX,nX), crossCov -----
__global__ __launch_bounds__(256) void build_kernel(const float* __restrict__ x,
                                                    const float* __restrict__ trainX,
                                                    const float* __restrict__ xn,
                                                    const float* __restrict__ tn,
                                                    const int* __restrict__ idxBuf,
                                                    const float* __restrict__ aP,
                                                    const float* __restrict__ lP,
                                                    float* __restrict__ autoBuf,
                                                    float* __restrict__ c0g) {
    constexpr int S = D + 4;
    __shared__ __align__(16) float nXs[128 * S];               // ~66 KB
    __shared__ float xq[128], nrm[128];
    __shared__ int sidx[128];
    const int b = blockIdx.x, tid = threadIdx.x;
    const int lane = tid & 31, wv = tid >> 5;
    const float av = *aP, lv = *lP;

    if (tid < 128) sidx[tid] = idxBuf[(size_t)b * K + tid];
    if (tid < 128) xq[tid] = x[(size_t)b * D + tid];
    __syncthreads();
    if (tid < 128) nrm[tid] = tn[sidx[tid]];
#if HAVE_ASYNC_LDS
    for (int q = tid; q < 128 * (D / 4); q += 256) {           // gather neighbor rows
        const int r = q >> 5, c4 = q & 31;
        const int src = sidx[r];
        async_load_b128(trainX + (size_t)src * D + c4 * 4, &nXs[r * S + c4 * 4]);
    }
    wait_async0();
#else
    for (int r = wv * 16; r < wv * 16 + 16; ++r) {
        const int src = sidx[r];
        float4 v = ((const float4*)(trainX + (size_t)src * D))[lane];
        *((float4*)&nXs[r * S + lane * 4]) = v;
    }
#endif
    __syncthreads();

    if (tid < 128) {                                           // crossCov
        float dot = 0.f;
        const float* rp = &nXs[tid * S];
        for (int d = 0; d < D; ++d) dot += xq[d] * rp[d];
        float d2 = fmaxf(xn[b] + nrm[tid] - 2.f * dot, 0.f);
        c0g[(size_t)b * K + tid] = av * expf(-sqrtf(d2) / lv);
    }
    __syncthreads();

    const int m = lane & 15, k0 = (lane >> 4) * 2;
    const int hi = lane >> 4, nn = lane & 15;
    (void)m; (void)k0; (void)hi; (void)nn;
    for (int tj = 0; tj < 8; ++tj) {                           // wave wv owns tile row wv
#if HAVE_WMMA_F32
        v8f acc = {0.f, 0.f, 0.f, 0.f, 0.f, 0.f, 0.f, 0.f};
        for (int s = 0; s < 32; ++s) {
            const int kk = 4 * s + k0;
            v2f a2 = *(const v2f*)&nXs[(wv * 16 + m) * S + kk];
            v2f b2 = *(const v2f*)&nXs[(tj * 16 + m) * S + kk];
            acc = __builtin_amdgcn_wmma_f32_16x16x4_f32(false, a2, false, b2,
                                                        (short)0, acc, false, false);
        }
#else
        float acc[8] = {0.f, 0.f, 0.f, 0.f, 0.f, 0.f, 0.f, 0.f};
        for (int k = 0; k < D; ++k) {
            float tv = nXs[(tj * 16 + nn) * S + k];
            #pragma unroll
            for (int vv = 0; vv < 8; ++vv) acc[vv] += nXs[(wv * 16 + vv + 8 * hi) * S + k] * tv;
        }
#endif
        #pragma unroll
        for (int vv = 0; vv < 8; ++vv) {
            const int i = wv * 16 + vv + 8 * hi;
            const int j = tj * 16 + nn;
            float d2 = fmaxf(nrm[i] + nrm[j] - 2.f * acc[vv], 0.f);
            autoBuf[((size_t)b << 14) + i * 128 + j] = av * expf(-sqrtf(d2) / lv);
        }
    }
}

// ------------- per-batch: Cholesky solve, y = w.nY + ym, yVar = a - w.c ----
__global__ __launch_bounds__(256) void solve_kernel(const float* __restrict__ autoBuf,
                                                    const float* __restrict__ c0g,
                                                    const int* __restrict__ idxBuf,
                                                    const float* __restrict__ trainy,
                                                    const float* __restrict__ ymean,
                                                    const float* __restrict__ noise,
                                                    const float* __restrict__ aP,
                                                    float* __restrict__ yOut,
                                                    float* __restrict__ yVar) {
    constexpr int S = D + 4;
    __shared__ __align__(16) float As[128 * S];                // ~66 KB
    __shared__ float nYs[128 * 32];                            // 16 KB
    __shared__ float cs[128], c0s[128];
    __shared__ int sidx[128];
    __shared__ float invd, red[256];
    const int b = blockIdx.x, tid = threadIdx.x;
    const float av = *aP;

#if HAVE_ASYNC_LDS
    for (int q = tid; q < 128 * (D / 4); q += 256) {
        const int i = q >> 5, c4 = q & 31;
        async_load_b128(autoBuf + ((size_t)b << 14) + i * 128 + c4 * 4,
                        &As[i * S + c4 * 4]);
    }
#else
    for (int e = tid; e < 128 * 128; e += 256) {
        const int i = e >> 7, j = e & 127;
        As[i * S + j] = autoBuf[((size_t)b << 14) + e];
    }
#endif
    if (tid < 128) {
        sidx[tid] = idxBuf[(size_t)b * K + tid];
        const float c = c0g[(size_t)b * K + tid];
        cs[tid] = c; c0s[tid] = c;
    }
    wait_async0();
    __syncthreads();
    if (tid < 128) {                                           // nY = trainy[idx] - ym + noise
        const int src = sidx[tid];
        for (int o = 0; o < O; ++o)
            nYs[tid * 32 + o] = trainy[(size_t)src * O + o] - ymean[b * O + o]
                              + noise[((size_t)b * K + tid) * O + o];
    }
    __syncthreads();

    for (int k = 0; k < 128; ++k) {                            // Cholesky (lower, in place)
        if (tid == 0) {
            const float dkk = sqrtf(fmaxf(As[k * S + k], 1e-30f));
            As[k * S + k] = dkk; invd = 1.f / dkk;
        }
        __syncthreads();
        for (int i = k + 1 + tid; i < 128; i += 256) As[i * S + k] *= invd;
        __syncthreads();
        for (int i = k + 1 + tid; i < 128; i += 256) {
            const float lik = As[i * S + k];
            for (int j = k + 1; j <= i; ++j) As[i * S + j] -= lik * As[j * S + k];
        }
        __syncthreads();
    }
    for (int k = 0; k < 128; ++k) {                            // L z = c
        if (tid == 0) cs[k] /= As[k * S + k];
        __syncthreads();
        const float zk = cs[k];
        for (int i = k + 1 + tid; i < 128; i += 256) cs[i] -= As[i * S + k] * zk;
        __syncthreads();
    }
    for (int k = 127; k >= 0; --k) {                           // L^T w = z
        if (tid == 0) cs[k] /= As[k * S + k];
        __syncthreads();
        const float wk = cs[k];
        for (int i = tid; i < k; i += 256) cs[i] -= As[k * S + i] * wk;
        __syncthreads();
    }

    if (tid < O) {
        float s = 0.f;
        for (int i = 0; i < 128; ++i) s += cs[i] * nYs[i * 32 + tid];
        yOut[b * O + tid] = ymean[b * O + tid] + s;
    }
    float p = 0.f;
    for (int i = tid; i < 128; i += 256) p += cs[i] * c0s[i];
    red[tid] = p;
    __syncthreads();
    if (tid == 0) {
        float s = 0.f;
        for (int t = 0; t < 256; ++t) s += red[t];
        yVar[b] = av - s;
    }
}

// --------------------------------------------------------------------------
extern "C" void kernel_launch(void* const* d_in, const int* in_sizes, int n_in,
                              void* d_out, int out_size, void* d_ws, size_t ws_size,
                              hipStream_t stream) {
    (void)in_sizes; (void)n_in; (void)out_size; (void)ws_size;
    const float* x      = (const float*)d_in[0];
    const float* trainX = (const float*)d_in[1];
    const float* trainy = (const float*)d_in[2];
    const float* W      = (const float*)d_in[3];
    const float* bias   = (const float*)d_in[4];
    const float* lP     = (const float*)d_in[5];
    const float* aP     = (const float*)d_in[6];
    const float* noise  = (const float*)d_in[7];
    float* out = (float*)d_out;

    char* ws = (char*)d_ws;
    size_t off = 0;
    auto alloc = [&](size_t bytes) -> void* {
        void* p = ws + off;
        off = (off + bytes + 255) & ~(size_t)255;
        return p;
    };
    float* ymean    = (float*)alloc((size_t)Bq * O * 4);
    float* tn       = (float*)alloc((size_t)N * 4);
    float* xn       = (float*)alloc((size_t)Bq * 4);
    int*   idxBuf   = (int*)  alloc((size_t)Bq * K * 4);
    float* c0g      = (float*)alloc((size_t)Bq * K * 4);
    float* autoBuf  = (float*)alloc((size_t)Bq * 128 * 128 * 4);   // 64 MB
    float* distSlab = (float*)alloc((size_t)SLAB * NPAD * 4);      // 51 MB, reused per slab

    norms_kernel<<<(N + 7) / 8, 256, 0, stream>>>(trainX, tn, N);
    norms_kernel<<<(Bq + 7) / 8, 256, 0, stream>>>(x, xn, Bq);
    ymean_kernel<<<Bq, 32, 0, stream>>>(x, W, bias, ymean);

    for (int slab = 0; slab < NSLAB; ++slab) {
        const int rowBase = slab * SLAB;
        dist_kernel<<<dim3(NPAD / 64, SLAB / 16), 128, 0, stream>>>(
            x, trainX, xn, tn, distSlab, rowBase);
        select_kernel<<<SLAB, 256, 0, stream>>>(distSlab, rowBase, idxBuf);
    }

    build_kernel<<<Bq, 256, 0, stream>>>(x, trainX, xn, tn, idxBuf, aP, lP, autoBuf, c0g);
    solve_kernel<<<Bq, 256, 0, stream>>>(autoBuf, c0g, idxBuf, trainy, ymean, noise, aP,
                                         out, out + (size_t)Bq * O);
}